// GCN_Net_61409442398221
// MI455X (gfx1250) — compile-verified
//
#include <hip/hip_runtime.h>
#include <hip/hip_bf16.h>

typedef float v2f __attribute__((ext_vector_type(2)));
typedef float v8f __attribute__((ext_vector_type(8)));

// ---------------- elementwise / degree kernels ----------------
__global__ void fill_kernel(float* __restrict__ p, float v, int n) {
  int i = blockIdx.x * blockDim.x + threadIdx.x;
  if (i < n) p[i] = v;
}

__global__ void count_deg_kernel(const int* __restrict__ dst, float* __restrict__ deg, int e) {
  int i = blockIdx.x * blockDim.x + threadIdx.x;
  if (i < e) atomicAdd(&deg[dst[i]], 1.0f);
}

__global__ void rsqrt_kernel(float* __restrict__ deg, int n) {
  int i = blockIdx.x * blockDim.x + threadIdx.x;
  if (i < n) deg[i] = rsqrtf(deg[i]);
}

// ---------------- fp32 WMMA GEMM: C[M,64] = A[M,K] @ B[K,64] ----------------
// K is compile-time (128 or 64): fully unrolled straight-line WMMA chain with
// immediate-offset global_load_b64 for A fragments. The 16-column B tile is
// staged in LDS once per block in a paired [K/2][16][2] layout so each lane's
// per-step B fragment is one ds_load_b64 with an immediate offset.
template <int K>
__global__ void gemm_wmma_f32_kernel(const float* __restrict__ A,
                                     const float* __restrict__ B,
                                     float* __restrict__ C, int M) {
  constexpr int Nc = 64;
  __shared__ float bsh[K * 16];

  const int lane  = threadIdx.x & 31;
  const int wave  = threadIdx.x >> 5;
  const int nbase = blockIdx.y * 16;

  // Stage B[:, nbase:nbase+16] into LDS: B[k][nbase+c] -> bsh[(k>>1)*32 + c*2 + (k&1)]
  for (int i = threadIdx.x; i < K * 16; i += blockDim.x) {
    int k = i >> 4, c = i & 15;
    bsh[(k >> 1) * 32 + c * 2 + (k & 1)] = B[(size_t)k * Nc + nbase + c];
  }
  __syncthreads();

  const int mtile = blockIdx.x * (blockDim.x >> 5) + wave;
  if (mtile * 16 >= M) return;                 // wave-uniform: EXEC all-1s inside
  const int mbase = mtile * 16;
  const int half  = lane >> 4;                 // 0: lanes 0-15 (K=0,1), 1: lanes 16-31 (K=2,3)
  const int l15   = lane & 15;
  int rowA = mbase + l15; if (rowA >= M) rowA = M - 1;   // tail clamp (read side only)

  const float* ap = A + (size_t)rowA * K + half * 2;     // per-lane base; +k is immediate
  const float* bp = bsh + l15 * 2 + half * 32;           // + (k/2)*32 is immediate

  v8f acc = {};
#pragma unroll
  for (int k = 0; k < K; k += 4) {
    v2f a = *(const v2f*)(ap + k);                  // global_load_b64, imm offset
    v2f b = *(const v2f*)(bp + (k >> 1) * 32);      // ds_load_b64, imm offset
    acc = __builtin_amdgcn_wmma_f32_16x16x4_f32(false, a, false, b,
                                                (short)0, acc, false, false);
  }

  const int rbase = mbase + half * 8;          // C layout: VGPR r -> row r / r+8
#pragma unroll
  for (int r = 0; r < 8; ++r) {
    int row = rbase + r;
    if (row < M) C[(size_t)row * Nc + nbase + l15] = acc[r];
  }
}

// ---------------- sparse aggregation: agg[dst] += norm * t[src] ----------------
// One wave32 per edge (plus N virtual self-loop edges); 2 features per lane.
__global__ void aggregate_kernel(const float* __restrict__ t,
                                 const int* __restrict__ src,
                                 const int* __restrict__ dst,
                                 const float* __restrict__ dinv,
                                 float* __restrict__ agg,
                                 int E, int Nn) {
  int idx  = blockIdx.x * blockDim.x + threadIdx.x;
  int e    = idx >> 5;                          // wave-uniform edge id
  int lane = idx & 31;
  int total = E + Nn;
  if (e >= total) return;
  int s, d;
  if (e < E) { s = src[e]; d = dst[e]; }
  else       { s = d = e - E; }                 // self loop
  float norm = dinv[s] * dinv[d];
  const float* ts = t   + (size_t)s * 64;
  float*       ad = agg + (size_t)d * 64;
  atomicAdd(&ad[lane],      ts[lane]      * norm);
  atomicAdd(&ad[lane + 32], ts[lane + 32] * norm);
}

__global__ void bias_act_kernel(const float* __restrict__ in, const float* __restrict__ bias,
                                float* __restrict__ out, int n64, int do_relu) {
  int i = blockIdx.x * blockDim.x + threadIdx.x;
  if (i < n64) {
    float v = in[i] + bias[i & 63];
    if (do_relu) v = fmaxf(v, 0.0f);
    out[i] = v;
  }
}

// ---------------- mean pool over graph ids ----------------
__global__ void pool_kernel(const float* __restrict__ h, const float* __restrict__ bias,
                            const int* __restrict__ batch,
                            float* __restrict__ sums, float* __restrict__ cnt, int Nn) {
  int idx  = blockIdx.x * blockDim.x + threadIdx.x;
  int node = idx >> 5;
  int lane = idx & 31;
  if (node >= Nn) return;
  int g = batch[node];
  const float* hv = h + (size_t)node * 64;
  atomicAdd(&sums[g * 64 + lane],      hv[lane]      + bias[lane]);
  atomicAdd(&sums[g * 64 + lane + 32], hv[lane + 32] + bias[lane + 32]);
  if (lane == 0) atomicAdd(&cnt[g], 1.0f);
}

__global__ void finalize_kernel(const float* __restrict__ sums, const float* __restrict__ cnt,
                                float* __restrict__ out, int n) {
  int i = blockIdx.x * blockDim.x + threadIdx.x;
  if (i < n) out[i] = sums[i] / fmaxf(cnt[i >> 6], 1.0f);
}

extern "C" void kernel_launch(void* const* d_in, const int* in_sizes, int n_in,
                              void* d_out, int out_size, void* d_ws, size_t ws_size,
                              hipStream_t stream) {
  const float* x    = (const float*)d_in[0];
  const int*   ei   = (const int*)d_in[1];
  const int*   batch= (const int*)d_in[2];
  const float* W1   = (const float*)d_in[3];
  const float* b1   = (const float*)d_in[4];
  const float* W2   = (const float*)d_in[5];
  const float* b2   = (const float*)d_in[6];
  float* out = (float*)d_out;

  const int Nn = in_sizes[0] / 128;   // 100000 nodes
  const int E  = in_sizes[1] / 2;     // 1600000 edges
  const int G  = out_size / 64;       // 128 graphs
  const int* src = ei;
  const int* dst = ei + E;

  // workspace carve-up (all float-typed, 8B aligned offsets)
  float* dinv = (float*)d_ws;                    // N
  float* bufA = dinv + Nn;                       // N*64
  float* bufB = bufA + (size_t)Nn * 64;          // N*64
  float* sums = bufB + (size_t)Nn * 64;          // G*64
  float* cnt  = sums + (size_t)G * 64;           // G (contiguous after sums)

  const int TPB = 256;
  auto nb = [](long long n, int tpb) { return (unsigned)((n + tpb - 1) / tpb); };

  // ---- symmetric-normalization degrees: deg = indeg + 1 (self loop); dinv = rsqrt ----
  fill_kernel     <<<nb(Nn, TPB), TPB, 0, stream>>>(dinv, 1.0f, Nn);
  count_deg_kernel<<<nb(E,  TPB), TPB, 0, stream>>>(dst, dinv, E);
  rsqrt_kernel    <<<nb(Nn, TPB), TPB, 0, stream>>>(dinv, Nn);

  const dim3 gemm_grid(nb((Nn + 15) / 16, 8), 4);   // 8 waves/block (one m-tile each), 4 n-tiles

  // ---- layer 1: t = x @ W1 ; agg ; h1 = relu(agg + b1) ----
  gemm_wmma_f32_kernel<128><<<gemm_grid, TPB, 0, stream>>>(x, W1, bufA, Nn);
  fill_kernel      <<<nb((long long)Nn * 64, TPB), TPB, 0, stream>>>(bufB, 0.0f, Nn * 64);
  aggregate_kernel <<<nb((long long)(E + Nn) * 32, TPB), TPB, 0, stream>>>(bufA, src, dst, dinv, bufB, E, Nn);
  bias_act_kernel  <<<nb((long long)Nn * 64, TPB), TPB, 0, stream>>>(bufB, b1, bufA, Nn * 64, 1);

  // ---- layer 2: t2 = h1 @ W2 ; agg ----
  gemm_wmma_f32_kernel<64><<<gemm_grid, TPB, 0, stream>>>(bufA, W2, bufB, Nn);
  fill_kernel      <<<nb((long long)Nn * 64, TPB), TPB, 0, stream>>>(bufA, 0.0f, Nn * 64);
  aggregate_kernel <<<nb((long long)(E + Nn) * 32, TPB), TPB, 0, stream>>>(bufB, src, dst, dinv, bufA, E, Nn);

  // ---- global mean pool (fuses +b2) ----
  fill_kernel    <<<nb(G * 64 + G, TPB), TPB, 0, stream>>>(sums, 0.0f, G * 64 + G);
  pool_kernel    <<<nb((long long)Nn * 32, TPB), TPB, 0, stream>>>(bufA, b2, batch, sums, cnt, Nn);
  finalize_kernel<<<nb(G * 64, TPB), TPB, 0, stream>>>(sums, cnt, out, G * 64);
}